// ExpMinProcessor_21036749816207
// MI455X (gfx1250) — compile-verified
//
#include <hip/hip_runtime.h>
#include <hip/hip_bf16.h>
#include <stdint.h>

// ---------------------------------------------------------------------------
// Top-p + exponential-minimum watermark sampler for MI455X (gfx1250).
//   Pass 1 (split 4x/row): online softmax stats (max, Z) + NEG_FILL fill (NT).
//   Pass 1b: combine per-slice (max, Z) partials.
//   Pass 2 (1 WG/row): 2-level histogram radix-select of top-p threshold L*.
//           Level 0 stages row tiles via the Tensor Data Mover
//           (tensor_load_to_lds, TENSORcnt); level 1 stages via per-lane
//           global_load_async_to_lds_b32 (ASYNCcnt). Both double-buffered.
//   Pass 3 (split 4x/row): argmin of log(-log(xi)) - logit over {logit >= L*}.
//   Pass 3b: combine per-slice argmin partials, write POS_FILL winner.
// HBM floor: 131MB read + 131MB write ~= 11.3us @ 23.3 TB/s; refinement
// passes re-read the row from the 192MB L2.
// ---------------------------------------------------------------------------

#define TOPP     0.9f
#define NEG_FILL -100000.0f
#define POS_FILL  100000.0f

constexpr int   V      = 128000;
constexpr int   V4     = V / 4;
constexpr int   BLK    = 1024;          // 32 waves (wave32) per workgroup
constexpr int   SPLIT  = 4;             // row slices for the HBM-bound passes
constexpr int   S4     = V4 / SPLIT;    // 8000 float4 per slice
constexpr int   NBINS  = 1024;
constexpr int   TILE   = 1024;          // floats per staged LDS tile
constexpr int   NTILES = V / TILE;      // 125
constexpr float RANGE1 = 16.0f;         // exp(-16)*128000 ~ 1.4e-2 << 0.1*Z

typedef float    fvec4 __attribute__((ext_vector_type(4)));
typedef unsigned u32x4 __attribute__((ext_vector_type(4)));
typedef int      i32x4 __attribute__((ext_vector_type(4)));
typedef int      i32x8 __attribute__((ext_vector_type(8)));

#if __has_builtin(__builtin_amdgcn_tensor_load_to_lds)
#define HAVE_TDM 1
#else
#define HAVE_TDM 0
#endif

// Low 32 bits of a generic LDS pointer are the wave-relative LDS byte offset
// (aperture lives in the high 32 bits per the CDNA5 flat-address mapping).
__device__ __forceinline__ unsigned lds_offset(const void* p) {
  return (unsigned)(uintptr_t)p;
}

// GLOBAL_LOAD_ASYNC_TO_LDS_B32 (GV mode): per-lane LDS dest in VDST, per-lane
// 64-bit global address in VADDR pair. Tracked with ASYNCcnt.
__device__ __forceinline__ void async_copy_f32(unsigned lds_byte, const float* g) {
  asm volatile("global_load_async_to_lds_b32 %0, %1, off"
               :: "v"(lds_byte), "v"((unsigned long long)(uintptr_t)g)
               : "memory");
}

// Tensor Data Mover: one instruction DMAs a [nelem x 1] f32 tile into LDS.
// D# layout per CDNA5 ISA.
__device__ __forceinline__ void tdm_load_tile(unsigned lds_byte, const float* g,
                                              unsigned nelem) {
#if HAVE_TDM
  unsigned long long ga = (unsigned long long)(uintptr_t)g;
  u32x4 g0;
  g0[0] = 1u;                                   // count=1 (valid user D#)
  g0[1] = lds_byte;                             // lds_addr (bytes)
  g0[2] = (unsigned)(ga & 0xFFFFFFFFu);         // global_addr[31:0]
  g0[3] = (unsigned)((ga >> 32) & 0x01FFFFFFu)  // global_addr[56:32]
          | 0x80000000u;                        // type=2 ("image")
  i32x8 g1;
  g1[0] = (int)(2u << 16);                      // data_size=2 -> 4 bytes
  g1[1] = (int)((nelem & 0xFFFFu) << 16);       // tensor_dim0[15:0]
  g1[2] = (int)((nelem >> 16) | (1u << 16));    // tensor_dim0[31:16], dim1=1
  g1[3] = (int)(nelem << 16);                   // tile_dim0 = nelem
  g1[4] = 1;                                    // tile_dim1=1, tile_dim2=0
  g1[5] = (int)nelem;                           // tensor_dim0_stride[31:0]
  g1[6] = 0;                                    // stride hi / dim1_stride lo
  g1[7] = 0;                                    // dim1_stride hi
  i32x4 z4 = {0, 0, 0, 0};                      // groups 2/3 unused (<=2D)
#if defined(__clang_major__) && (__clang_major__ >= 23)
  i32x8 z8 = {0, 0, 0, 0, 0, 0, 0, 0};
  __builtin_amdgcn_tensor_load_to_lds(g0, g1, z4, z4, z8, 0);
#else
  __builtin_amdgcn_tensor_load_to_lds(g0, g1, z4, z4, 0);
#endif
#else
  (void)lds_byte; (void)g; (void)nelem;
#endif
}

// ---------------------------------------------------------------------------
// Pass 1: per-slice max & sum(exp) (online); fill out slice = NEG_FILL (NT).
// grid = (SPLIT, B)
// ---------------------------------------------------------------------------
__global__ void __launch_bounds__(BLK)
rowstats_partial_kernel(const float* __restrict__ logits, float* __restrict__ out,
                        float* __restrict__ pmax, float* __restrict__ psum) {
  const int slice = blockIdx.x;
  const int row   = blockIdx.y;
  const int tid   = threadIdx.x;
  const fvec4* __restrict__ l4 = reinterpret_cast<const fvec4*>(logits + (size_t)row * V);
  fvec4* __restrict__ o4 = reinterpret_cast<fvec4*>(out + (size_t)row * V);
  const fvec4 neg4 = {NEG_FILL, NEG_FILL, NEG_FILL, NEG_FILL};

  float m = -__builtin_inff();
  float s = 0.0f;
  const int i0 = slice * S4, i1 = i0 + S4;
  for (int i = i0 + tid; i < i1; i += BLK) {
    fvec4 x = l4[i];
    __builtin_nontemporal_store(neg4, &o4[i]);   // streaming fill, no L2 keep
#pragma unroll
    for (int c = 0; c < 4; ++c) {
      float xc = x[c];
      if (xc > m) { s = s * __expf(m - xc) + 1.0f; m = xc; }
      else        { s += __expf(xc - m); }
    }
  }
#pragma unroll
  for (int o = 16; o > 0; o >>= 1) {
    float om = __shfl_xor(m, o, 32);
    float os = __shfl_xor(s, o, 32);
    float M  = fmaxf(m, om);
    s = s * __expf(m - M) + os * __expf(om - M);
    m = M;
  }
  __shared__ float rm[32], rs[32];
  if ((tid & 31) == 0) { rm[tid >> 5] = m; rs[tid >> 5] = s; }
  __syncthreads();
  if (tid < 32) {
    m = rm[tid]; s = rs[tid];
#pragma unroll
    for (int o = 16; o > 0; o >>= 1) {
      float om = __shfl_xor(m, o, 32);
      float os = __shfl_xor(s, o, 32);
      float M  = fmaxf(m, om);
      s = s * __expf(m - M) + os * __expf(om - M);
      m = M;
    }
    if (tid == 0) { pmax[row * SPLIT + slice] = m; psum[row * SPLIT + slice] = s; }
  }
}

__global__ void combine_stats_kernel(const float* __restrict__ pmax,
                                     const float* __restrict__ psum,
                                     float* __restrict__ ws_max,
                                     float* __restrict__ ws_z) {
  const int row = blockIdx.x;
  if (threadIdx.x == 0) {
    float m = pmax[row * SPLIT], s = psum[row * SPLIT];
    for (int k = 1; k < SPLIT; ++k) {
      float om = pmax[row * SPLIT + k], os = psum[row * SPLIT + k];
      float M  = fmaxf(m, om);
      s = s * __expf(m - M) + os * __expf(om - M);
      m = M;
    }
    ws_max[row] = m; ws_z[row] = s;
  }
}

// ---------------------------------------------------------------------------
// Pass 2: two-level histogram select of L*. Level 0 staged by the TDM,
// level 1 staged by per-lane async copies; both double-buffered.
// ---------------------------------------------------------------------------
__global__ void __launch_bounds__(BLK)
threshold_kernel(const float* __restrict__ logits, const float* __restrict__ ws_max,
                 const float* __restrict__ ws_z, float* __restrict__ ws_thr) {
  const int row = blockIdx.x;
  const int tid = threadIdx.x;
  const float* __restrict__ lrow = logits + (size_t)row * V;

  __shared__ float hist[NBINS];
  __shared__ float csum[NBINS / 4];
  __shared__ float tile[2][TILE];
  __shared__ float sh_lo, sh_w, sh_abv;
  __shared__ int   sh_found;

  const float rowmax = ws_max[row];
  const float target = TOPP * ws_z[row];

  if (tid == 0) {
    sh_lo    = rowmax - RANGE1;
    sh_w     = RANGE1 / (float)NBINS;
    sh_abv   = 0.0f;
    sh_found = 1;
  }
  __syncthreads();

  const unsigned lbase = lds_offset(&tile[0][0]);
  const int wave = tid >> 5;

  for (int level = 0; level < 2; ++level) {
    const float lo   = sh_lo;
    const float w    = sh_w;
    const float abv0 = sh_abv;
    const int   go   = sh_found;            // uniform across block
    __syncthreads();
    if (go) {
      const float hi   = lo + w * (float)NBINS;
      const float invw = 1.0f / w;
      hist[tid] = 0.0f;
      __syncthreads();

      auto accum = [&](float x) {
        bool in = (level == 0) ? (x >= lo) : (x >= lo && x < hi);
        if (in) {
          int b = (int)((x - lo) * invw);
          b = b < 0 ? 0 : (b > NBINS - 1 ? NBINS - 1 : b);
          atomicAdd(&hist[b], __expf(x - rowmax));   // ds_add_f32
        }
      };

      int cur = 0;
      if (HAVE_TDM && level == 0) {
        // ---- Tensor Data Mover staging: 1 instruction per 4KB tile ----
        if (wave == 0) tdm_load_tile(lbase, lrow, TILE);
        for (int t = 0; t < NTILES; ++t) {
          if (wave == 0) {
            if (t + 1 < NTILES) {
              tdm_load_tile(lbase + 4u * (unsigned)((cur ^ 1) * TILE),
                            lrow + (size_t)(t + 1) * TILE, TILE);
              __builtin_amdgcn_s_wait_tensorcnt((short)1);  // oldest tile done
            } else {
              __builtin_amdgcn_s_wait_tensorcnt((short)0);
            }
          }
          __syncthreads();                  // publish tile t to all waves
          accum(tile[cur][tid]);
          __syncthreads();                  // reads done before buffer reuse
          cur ^= 1;
        }
      } else {
        // ---- per-lane async Global->LDS staging (ASYNCcnt) ----
        async_copy_f32(lbase + 4u * tid, lrow + tid);
        for (int t = 0; t < NTILES; ++t) {
          if (t + 1 < NTILES) {
            async_copy_f32(lbase + 4u * ((unsigned)((cur ^ 1) * TILE) + tid),
                           lrow + (size_t)(t + 1) * TILE + tid);
            asm volatile("s_wait_asynccnt 1" ::: "memory");
          } else {
            asm volatile("s_wait_asynccnt 0" ::: "memory");
          }
          __syncthreads();
          accum(tile[cur][tid]);
          __syncthreads();
          cur ^= 1;
        }
      }

      // suffix-sum scan from the top bin down to find the crossing bin.
      if (tid < NBINS / 4) {
        csum[tid] = hist[4 * tid] + hist[4 * tid + 1] +
                    hist[4 * tid + 2] + hist[4 * tid + 3];
      }
      __syncthreads();
      if (tid == 0) {
        float run = abv0;                   // weight already above this range
        int   bstar = -1;
        float above_b = run;
        for (int c = NBINS / 4 - 1; c >= 0; --c) {
          if (run + csum[c] >= target) {
            for (int b = 4 * c + 3; b >= 4 * c; --b) {
              if (run + hist[b] >= target) { bstar = b; above_b = run; break; }
              run += hist[b];
            }
            if (bstar < 0) { bstar = 4 * c; above_b = run; }  // fp-assoc safety
            break;
          }
          run += csum[c];
        }
        if (bstar < 0) {
          sh_found = 0;                     // crossing below range: keep all
        } else {
          sh_found = 1;
          sh_lo  = lo + w * (float)bstar;   // refined lower edge
          sh_w   = w * (1.0f / (float)NBINS);
          sh_abv = above_b;
        }
      }
    }
    __syncthreads();
  }

  if (tid == 0) ws_thr[row] = sh_found ? sh_lo : -3.0e38f;
}

// ---------------------------------------------------------------------------
// Pass 3: per-slice argmin of key = log(-log(xi)) - logit over kept set.
// (Monotone transform of -log(xi)/p: p = exp(logit-max)/Z, Z & max constant.)
// grid = (SPLIT, B)
// ---------------------------------------------------------------------------
__global__ void __launch_bounds__(BLK)
argmin_partial_kernel(const float* __restrict__ logits, const float* __restrict__ xi,
                      const float* __restrict__ ws_thr,
                      float* __restrict__ pval, int* __restrict__ pidx) {
  const int slice = blockIdx.x;
  const int row   = blockIdx.y;
  const int tid   = threadIdx.x;
  const fvec4* __restrict__ l4 = reinterpret_cast<const fvec4*>(logits + (size_t)row * V);
  const fvec4* __restrict__ x4 = reinterpret_cast<const fvec4*>(xi);
  const float thr = ws_thr[row];

  float best = __builtin_inff();
  int   bidx = 0;
  const int i0 = slice * S4, i1 = i0 + S4;
  for (int i = i0 + tid; i < i1; i += BLK) {
    if (i + BLK < i1) {
      __builtin_prefetch(&l4[i + BLK], 0, 0);   // global_prefetch_b8
      __builtin_prefetch(&x4[i + BLK], 0, 0);
    }
    fvec4 lg = l4[i];
    fvec4 u  = x4[i];
#pragma unroll
    for (int c = 0; c < 4; ++c) {
      float x = lg[c];
      if (x >= thr) {
        float key = __logf(-__logf(u[c])) - x;
        if (key < best) { best = key; bidx = 4 * i + c; }
      }
    }
  }
#pragma unroll
  for (int o = 16; o > 0; o >>= 1) {
    float ov = __shfl_xor(best, o, 32);
    int   oi = __shfl_xor(bidx, o, 32);
    if (ov < best || (ov == best && oi < bidx)) { best = ov; bidx = oi; }
  }
  __shared__ float rv[32];
  __shared__ int   ri[32];
  if ((tid & 31) == 0) { rv[tid >> 5] = best; ri[tid >> 5] = bidx; }
  __syncthreads();
  if (tid < 32) {
    best = rv[tid]; bidx = ri[tid];
#pragma unroll
    for (int o = 16; o > 0; o >>= 1) {
      float ov = __shfl_xor(best, o, 32);
      int   oi = __shfl_xor(bidx, o, 32);
      if (ov < best || (ov == best && oi < bidx)) { best = ov; bidx = oi; }
    }
    if (tid == 0) { pval[row * SPLIT + slice] = best; pidx[row * SPLIT + slice] = bidx; }
  }
}

__global__ void combine_argmin_kernel(const float* __restrict__ pval,
                                      const int* __restrict__ pidx,
                                      float* __restrict__ out) {
  const int row = blockIdx.x;
  if (threadIdx.x == 0) {
    float best = pval[row * SPLIT];
    int   bidx = pidx[row * SPLIT];
    for (int k = 1; k < SPLIT; ++k) {
      float ov = pval[row * SPLIT + k];
      int   oi = pidx[row * SPLIT + k];
      if (ov < best || (ov == best && oi < bidx)) { best = ov; bidx = oi; }
    }
    out[(size_t)row * V + bidx] = POS_FILL;
  }
}

// ---------------------------------------------------------------------------
extern "C" void kernel_launch(void* const* d_in, const int* in_sizes, int n_in,
                              void* d_out, int out_size, void* d_ws, size_t ws_size,
                              hipStream_t stream) {
  (void)n_in; (void)out_size; (void)ws_size;
  // d_in[0] = input_ids (int64, unused by the reference computation)
  const float* logits = (const float*)d_in[1];
  const float* xi     = (const float*)d_in[2];
  float* out = (float*)d_out;
  const int B = in_sizes[1] / V;   // 256

  float* wsf    = (float*)d_ws;
  float* ws_max = wsf;                      // [B]
  float* ws_z   = wsf + B;                  // [B]
  float* ws_thr = wsf + 2 * B;              // [B]
  float* pmax   = wsf + 3 * B;              // [B*SPLIT]
  float* psum   = pmax + B * SPLIT;         // [B*SPLIT]
  float* pval   = psum + B * SPLIT;         // [B*SPLIT]
  int*   pidx   = (int*)(pval + B * SPLIT); // [B*SPLIT]

  rowstats_partial_kernel<<<dim3(SPLIT, B), BLK, 0, stream>>>(logits, out, pmax, psum);
  combine_stats_kernel<<<B, 32, 0, stream>>>(pmax, psum, ws_max, ws_z);
  threshold_kernel<<<B, BLK, 0, stream>>>(logits, ws_max, ws_z, ws_thr);
  argmin_partial_kernel<<<dim3(SPLIT, B), BLK, 0, stream>>>(logits, xi, ws_thr, pval, pidx);
  combine_argmin_kernel<<<B, 32, 0, stream>>>(pval, pidx, out);
}